// Detector_Jingzhui_84421877170821
// MI455X (gfx1250) — compile-verified
//
#include <hip/hip_runtime.h>
#include <cstdint>
#include <cstddef>

// ---------------- problem constants ----------------
#define THRESH   0.5f
#define IOU_THR  0.3f
#define TOPK     512
#define NCLS     80

#define OFF26    16224      // 32*13*13*3
#define OFF52    81120      // + 32*26*26*3
#define NTOT     340704     // + 32*52*52*3

#define NBIN     4096
#define CAP      4096

typedef __attribute__((ext_vector_type(2))) float v2f;
typedef __attribute__((ext_vector_type(8))) float v8f;

__constant__ float d_anch[18] = {
    116.f, 90.f, 156.f, 198.f, 373.f, 326.f,   // stride 32 (13x13)
     30.f, 61.f,  62.f,  45.f,  59.f, 119.f,   // stride 16 (26x26)
     10.f, 13.f,  16.f,  30.f,  33.f,  23.f    // stride  8 (52x52)
};

// ---------------- init: zero histogram + counter ----------------
__global__ void k_init(unsigned* __restrict__ hist, unsigned* __restrict__ counter) {
    int i = blockIdx.x * blockDim.x + threadIdx.x;
    if (i < NBIN) hist[i] = 0u;
    if (i == 0) counter[0] = 0u;
}

// ---------------- decode: 1 thread per box, fused histogram ----------------
__global__ void k_decode(const float* __restrict__ in13, const float* __restrict__ in26,
                         const float* __restrict__ in52, float* __restrict__ boxes,
                         float* __restrict__ scores, unsigned* __restrict__ hist) {
    int gid = blockIdx.x * blockDim.x + threadIdx.x;
    if (gid >= NTOT) return;

    const float* src; int l, W, anchOff; float stride;
    if (gid < OFF26)      { l = gid;         W = 13; stride = 32.f; src = in13; anchOff = 0;  }
    else if (gid < OFF52) { l = gid - OFF26; W = 26; stride = 16.f; src = in26; anchOff = 6;  }
    else                  { l = gid - OFF52; W = 52; stride =  8.f; src = in52; anchOff = 12; }

    // flatten order matches reshape(B,H,W,3,-1): idx = ((b*H+y)*W+x)*3+a
    int a = l % 3; int t = l / 3;
    int x = t % W; t /= W;
    int y = t % W; int b = t / W;
    int HW = W * W;

    const float* p = src + (size_t)(b * 255 + a * 85) * HW + (size_t)y * W + x;
    float o0 = p[0];
    float o1 = p[(size_t)1 * HW];
    float o2 = p[(size_t)2 * HW];
    float o3 = p[(size_t)3 * HW];
    float o4 = p[(size_t)4 * HW];

    // argmax over 80 class logits (first max wins, like jnp.argmax)
    float best = p[(size_t)5 * HW]; int bi = 0;
    for (int c = 1; c < NCLS; ++c) {
        __builtin_prefetch(p + (size_t)(5 + c + 8) * HW, 0, 0);
        float v = p[(size_t)(5 + c) * HW];
        if (v > best) { best = v; bi = c; }
    }

    float conf = 1.f / (1.f + expf(-o4));
    float sx   = 1.f / (1.f + expf(-o0));
    float sy   = 1.f / (1.f + expf(-o1));
    float cx = ((float)x + sx) * stride;
    float cy = ((float)y + sy) * stride;
    float w  = d_anch[anchOff + a * 2 + 0] * expf(o2);
    float h  = d_anch[anchOff + a * 2 + 1] * expf(o3);

    float* r = boxes + (size_t)gid * 7;
    r[0] = conf;
    r[1] = cx - 0.5f * w; r[2] = cy - 0.5f * h;
    r[3] = cx + 0.5f * w; r[4] = cy + 0.5f * h;
    r[5] = (float)bi;     r[6] = (float)b;

    float sc = (conf > THRESH) ? conf : 0.f;
    scores[gid] = sc;
    if (conf > THRESH) {
        // conf in (0.5,1): bits in (0x3F000000,0x3F800000); top 12 mantissa bits
        // are a monotonic 4096-bin key
        unsigned bits = __float_as_uint(conf);
        atomicAdd(&hist[(bits >> 11) & (NBIN - 1)], 1u);
    }
}

// ---------------- radix-select pivot bin (serial scan of 4096 bins) ----------------
__global__ void k_pivot(const unsigned* __restrict__ hist, unsigned* __restrict__ pivot) {
    if (threadIdx.x != 0 || blockIdx.x != 0) return;
    unsigned total = 0, pb = 0;
    for (int b = NBIN - 1; b >= 0; --b) {
        total += hist[b];
        if (total >= TOPK) { pb = (unsigned)b; break; }
    }
    pivot[0] = pb;   // if total < TOPK overall, pb==0 -> take everything above THRESH
}

// ---------------- compact candidates with bin >= pivot ----------------
__global__ void k_compact(const float* __restrict__ scores, const unsigned* __restrict__ pivot,
                          unsigned* __restrict__ counter, float* __restrict__ candScore,
                          int* __restrict__ candIdx) {
    int gid = blockIdx.x * blockDim.x + threadIdx.x;
    if (gid >= NTOT) return;
    float s = scores[gid];
    if (s > THRESH) {
        unsigned bin = (__float_as_uint(s) >> 11) & (NBIN - 1);
        if (bin >= pivot[0]) {
            unsigned pos = atomicAdd(counter, 1u);
            if (pos < CAP) { candScore[pos] = s; candIdx[pos] = gid; }
        }
    }
}

// ---------------- one-block bitonic sort of <=4096 candidates ----------------
// key = (~scoreBits << 32) | idx  -> ascending sort == score desc, idx asc (jax tie-break)
__global__ __launch_bounds__(1024) void k_sort(const float* __restrict__ candScore,
                                               const int* __restrict__ candIdx,
                                               const unsigned* __restrict__ counter,
                                               unsigned* __restrict__ topBits,
                                               int* __restrict__ topIdx) {
    __shared__ unsigned long long key[CAP];
    int tid = threadIdx.x;
    unsigned cnt = counter[0]; if (cnt > CAP) cnt = CAP;
    for (int i = tid; i < CAP; i += 1024) {
        unsigned long long k;
        if (i < (int)cnt) {
            unsigned sb = __float_as_uint(candScore[i]);
            k = ((unsigned long long)(~sb) << 32) | (unsigned)candIdx[i];
        } else {
            k = ~0ULL;   // pad: sorts last
        }
        key[i] = k;
    }
    __syncthreads();
    for (int kk = 2; kk <= CAP; kk <<= 1) {
        for (int j = kk >> 1; j > 0; j >>= 1) {
            for (int i = tid; i < CAP; i += 1024) {
                int ixj = i ^ j;
                if (ixj > i) {
                    unsigned long long a = key[i], b = key[ixj];
                    bool up = ((i & kk) == 0);
                    if (up ? (a > b) : (a < b)) { key[i] = b; key[ixj] = a; }
                }
            }
            __syncthreads();
        }
    }
    for (int i = tid; i < TOPK; i += 1024) {
        topBits[i] = ~(unsigned)(key[i] >> 32);
        topIdx[i]  = (int)(unsigned)(key[i] & 0xFFFFFFFFull);
    }
}

// ---------------- gather top-512 rows + areas + valid flags ----------------
__global__ __launch_bounds__(TOPK) void k_gather(const float* __restrict__ boxes,
                                                 const unsigned* __restrict__ topBits,
                                                 const int* __restrict__ topIdx,
                                                 float* __restrict__ candBox,
                                                 float* __restrict__ area,
                                                 int* __restrict__ validF) {
    int tid = threadIdx.x;
    float sc = __uint_as_float(topBits[tid]);
    int valid = (sc > THRESH) ? 1 : 0;
    float row[7] = {0.f, 0.f, 0.f, 0.f, 0.f, 0.f, 0.f};
    if (valid) {
        int idx = topIdx[tid];
        if ((unsigned)idx < (unsigned)NTOT) {
            const float* r = boxes + (size_t)idx * 7;
#pragma unroll
            for (int f = 0; f < 7; ++f) row[f] = r[f];
        } else {
            valid = 0;
        }
    }
#pragma unroll
    for (int f = 0; f < 7; ++f) candBox[tid * 7 + f] = row[f];
    float w = row[3] - row[1]; if (w < 0.f) w = 0.f;
    float h = row[4] - row[2]; if (h < 0.f) h = 0.f;
    area[tid]   = w * h;
    validF[tid] = valid;
}

// ---------------- IoU matrix: one wave per 16x16 tile ----------------
// sumArea tile (area_i + area_j) is a rank-2 GEMM: [area_i,1] x [[1..1],[area_j]]
// computed with V_WMMA_F32_16X16X4_F32; intersection on VALU; iou -> global.
__global__ __launch_bounds__(32) void k_iou(const float* __restrict__ candBox,
                                            const float* __restrict__ area,
                                            float* __restrict__ iou) {
    int tile = blockIdx.x;
    int ti = tile >> 5, tj = tile & 31;
    int r0 = ti * 16, c0 = tj * 16;
    int l  = (int)threadIdx.x;
    int lm = l & 15;
    bool lo = (l < 16);

    __shared__ float R[16][4], C[16][4];
    {
        int src_row = lo ? (r0 + lm) : (c0 + lm);
        const float* src = candBox + (size_t)src_row * 7 + 1;   // x1,y1,x2,y2
        float* dst = lo ? &R[lm][0] : &C[lm][0];
        dst[0] = src[0]; dst[1] = src[1]; dst[2] = src[2]; dst[3] = src[3];
    }

    // A (16x4 f32, 2 VGPRs): lane<16 holds K0=area_i, K1=1 for row M=lane;
    // lanes 16-31 hold K2=K3=0.  B (4x16): lane<16 holds K0=1, K1=area_j.
    float av = area[r0 + lm];
    float bv = area[c0 + lm];
    v2f A; A.x = lo ? av  : 0.f; A.y = lo ? 1.f : 0.f;
    v2f B; B.x = lo ? 1.f : 0.f; B.y = lo ? bv  : 0.f;
    v8f Csum = {0.f, 0.f, 0.f, 0.f, 0.f, 0.f, 0.f, 0.f};
    Csum = __builtin_amdgcn_wmma_f32_16x16x4_f32(
        /*neg_a=*/false, A, /*neg_b=*/false, B,
        /*c_mod=*/(short)0, Csum, /*reuse_a=*/false, /*reuse_b=*/false);

    __syncthreads();

    int rh = (l >> 4) << 3;     // D layout: VGPR v -> row v (lanes 0-15), v+8 (lanes 16-31)
#pragma unroll
    for (int v = 0; v < 8; ++v) {
        int r = v + rh;
        int c = lm;
        float xx1 = fmaxf(R[r][0], C[c][0]);
        float yy1 = fmaxf(R[r][1], C[c][1]);
        float xx2 = fminf(R[r][2], C[c][2]);
        float yy2 = fminf(R[r][3], C[c][3]);
        float iw = xx2 - xx1; if (iw < 0.f) iw = 0.f;
        float ih = yy2 - yy1; if (ih < 0.f) ih = 0.f;
        float inter = iw * ih;
        float u = inter / (Csum[v] - inter + 1e-9f);
        iou[(size_t)(r0 + r) * TOPK + (c0 + c)] = u;
    }
}

// ---------------- sequential greedy NMS + masked output ----------------
__global__ __launch_bounds__(TOPK) void k_nms(const float* __restrict__ iou,
                                              const int* __restrict__ validF,
                                              const float* __restrict__ candBox,
                                              float* __restrict__ out) {
    __shared__ int keep[TOPK];
    int tid = threadIdx.x;
    keep[tid] = validF[tid];
    __syncthreads();
    for (int i = 0; i < TOPK; ++i) {
        if (keep[i] && tid > i && keep[tid]) {
            if (iou[(size_t)i * TOPK + tid] > IOU_THR) keep[tid] = 0;
        }
        __syncthreads();
    }
    float m = keep[tid] ? 1.f : 0.f;
#pragma unroll
    for (int f = 0; f < 7; ++f) out[tid * 7 + f] = candBox[tid * 7 + f] * m;
}

// ---------------- host launcher ----------------
extern "C" void kernel_launch(void* const* d_in, const int* in_sizes, int n_in,
                              void* d_out, int out_size, void* d_ws, size_t ws_size,
                              hipStream_t stream) {
    (void)in_sizes; (void)n_in; (void)out_size; (void)ws_size;
    const float* in13 = (const float*)d_in[0];
    const float* in26 = (const float*)d_in[1];
    const float* in52 = (const float*)d_in[2];
    float* out = (float*)d_out;

    char* w = (char*)d_ws;
    size_t off = 0;
    auto alloc = [&](size_t bytes) -> void* {
        void* p = (void*)(w + off);
        off += bytes;
        off = (off + 255) & ~(size_t)255;
        return p;
    };
    float*    boxes     = (float*)   alloc((size_t)NTOT * 7 * sizeof(float));
    float*    scores    = (float*)   alloc((size_t)NTOT * sizeof(float));
    unsigned* hist      = (unsigned*)alloc(NBIN * sizeof(unsigned));
    unsigned* counter   = (unsigned*)alloc(sizeof(unsigned));
    unsigned* pivot     = (unsigned*)alloc(sizeof(unsigned));
    float*    candScore = (float*)   alloc(CAP * sizeof(float));
    int*      candIdx   = (int*)     alloc(CAP * sizeof(int));
    unsigned* topBits   = (unsigned*)alloc(TOPK * sizeof(unsigned));
    int*      topIdx    = (int*)     alloc(TOPK * sizeof(int));
    float*    candBox   = (float*)   alloc(TOPK * 7 * sizeof(float));
    float*    areaBuf   = (float*)   alloc(TOPK * sizeof(float));
    int*      validF    = (int*)     alloc(TOPK * sizeof(int));
    float*    iouM      = (float*)   alloc((size_t)TOPK * TOPK * sizeof(float));

    k_init   <<<(NBIN + 255) / 256, 256, 0, stream>>>(hist, counter);
    k_decode <<<(NTOT + 255) / 256, 256, 0, stream>>>(in13, in26, in52, boxes, scores, hist);
    k_pivot  <<<1, 1, 0, stream>>>(hist, pivot);
    k_compact<<<(NTOT + 255) / 256, 256, 0, stream>>>(scores, pivot, counter, candScore, candIdx);
    k_sort   <<<1, 1024, 0, stream>>>(candScore, candIdx, counter, topBits, topIdx);
    k_gather <<<1, TOPK, 0, stream>>>(boxes, topBits, topIdx, candBox, areaBuf, validF);
    k_iou    <<<(TOPK / 16) * (TOPK / 16), 32, 0, stream>>>(candBox, areaBuf, iouM);
    k_nms    <<<1, TOPK, 0, stream>>>(iouM, validF, candBox, out);
}